// NCDEAttention_49426483642795
// MI455X (gfx1250) — compile-verified
//
#include <hip/hip_runtime.h>
#include <math.h>

// Problem constants (match reference)
#define BDIM  4
#define SEQ   40
#define DMODEL 256
#define NH    8
#define DK    32
#define DV    32
#define INTERN 32
#define BH    32      // NH * BDIM
#define GSEG  32      // nc * (NQ-1) = 8*4
#define LD    33      // padded LDS row stride (A-operand bank-conflict avoidance)

typedef __attribute__((ext_vector_type(2))) float v2f;
typedef __attribute__((ext_vector_type(8))) float v8f;

// ---------------------------------------------------------------------------
// Generic f32 WMMA GEMM: C(160 x N) = A(160 x K) @ W(K x N), one wave per
// 16x16 tile, K-loop of V_WMMA_F32_16X16X4_F32 steps. mode==1 scatters the
// result into per-head (bh, s, c) layout used by qp/kp/vp.
// ---------------------------------------------------------------------------
__global__ __launch_bounds__(32)
void NCDE_wmma_gemm(const float* __restrict__ A, const float* __restrict__ W,
                    float* __restrict__ C, int K, int N, int mode)
{
  const int lane = threadIdx.x;
  const int half = lane >> 4;     // 0: lanes 0-15, 1: lanes 16-31
  const int l15  = lane & 15;
  const int mt = blockIdx.x, nt = blockIdx.y;
  const int row  = mt * 16 + l15;
  const int col  = nt * 16 + l15;
  const int koff = half * 2;      // A/B K-offset per ISA 16x4 / 4x16 layout

  v8f acc = {0.f, 0.f, 0.f, 0.f, 0.f, 0.f, 0.f, 0.f};
  for (int kk = 0; kk < (K >> 2); ++kk) {
    const int kb = (kk << 2) + koff;
    v2f a, b;
    a.x = A[row * K + kb];
    a.y = A[row * K + kb + 1];
    b.x = W[kb * N + col];
    b.y = W[(kb + 1) * N + col];
    acc = __builtin_amdgcn_wmma_f32_16x16x4_f32(false, a, false, b,
                                                (short)0, acc, false, false);
  }
  for (int r = 0; r < 8; ++r) {
    const int orow = mt * 16 + r + half * 8;   // C layout: VGPR r -> M=r / r+8
    const float v = acc[r];
    if (mode == 0) {
      C[orow * N + col] = v;
    } else {
      const int b_ = orow / SEQ, s = orow % SEQ;
      const int h = col >> 5, c = col & 31;
      const int bh = h * BDIM + b_;
      C[(bh * SEQ + s) * DK + c] = v;
    }
  }
}

// ---------------------------------------------------------------------------
// Natural cubic spline: Thomas algorithm, one thread per (bh, c) column.
// Coeffs stored time-major: coef[t * 1024 + bh*32 + c], t in [0, SEQ-2].
// ---------------------------------------------------------------------------
__global__ __launch_bounds__(256)
void NCDE_spline(const float* __restrict__ qp, const float* __restrict__ times,
                 float* __restrict__ ca, float* __restrict__ cb,
                 float* __restrict__ cc, float* __restrict__ cd)
{
  const int tid = blockIdx.x * 256 + threadIdx.x;   // 0..1023
  const int bh = tid >> 5, c = tid & 31;

  float kn[SEQ], y[SEQ];
  for (int t = 0; t < SEQ; ++t) {
    kn[t] = times[t];                       // t1d = times[0, :SEQ]
    y[t]  = qp[(bh * SEQ + t) * DK + c];
  }
  float h[SEQ - 1], sl[SEQ - 1];
  for (int i = 0; i < SEQ - 1; ++i) {
    h[i]  = kn[i + 1] - kn[i];
    sl[i] = (y[i + 1] - y[i]) / h[i];
  }
  // Tridiagonal: row i (0..37): sub=h[i], diag=2(h[i]+h[i+1]), sup=h[i+1]
  float cp[SEQ - 2], dp[SEQ - 2];
  {
    const float diag = 2.f * (h[0] + h[1]);
    cp[0] = h[1] / diag;
    dp[0] = 6.f * (sl[1] - sl[0]) / diag;
  }
  for (int i = 1; i < SEQ - 2; ++i) {
    const float denom = 2.f * (h[i] + h[i + 1]) - h[i] * cp[i - 1];
    cp[i] = h[i + 1] / denom;
    dp[i] = (6.f * (sl[i + 1] - sl[i]) - h[i] * dp[i - 1]) / denom;
  }
  float m[SEQ];
  m[0] = 0.f; m[SEQ - 1] = 0.f;
  m[SEQ - 2] = dp[SEQ - 3];
  for (int i = SEQ - 4; i >= 0; --i) m[i + 1] = dp[i] - cp[i] * m[i + 2];

  for (int t = 0; t < SEQ - 1; ++t) {
    const float A_ = y[t];
    const float B_ = sl[t] - h[t] * (2.f * m[t] + m[t + 1]) * (1.f / 6.f);
    const float C_ = m[t] * 0.5f;
    const float D_ = (m[t + 1] - m[t]) / (6.f * h[t]);
    const int o = t * (BH * DK) + tid;
    ca[o] = A_; cb[o] = B_; cc[o] = C_; cd[o] = D_;
  }
}

// ---------------------------------------------------------------------------
// Wave-level 40x32x32 GEMM (A in LDS, stride LD; B in LDS, stride ldb) via
// V_WMMA_F32_16X16X4_F32. A rows padded to 48 (pad rows only pollute unused
// output rows 40..47, which are never consumed).
// ---------------------------------------------------------------------------
__device__ __forceinline__
void wave_gemm_lds(const float* As, const float* Bs, int ldb, float* Ds,
                   const float* addv, bool dotanh, int lane)
{
  const int half = lane >> 4, l15 = lane & 15, koff = half * 2;
  for (int mt = 0; mt < 3; ++mt)
    for (int nt = 0; nt < 2; ++nt) {
      v8f acc = {0.f, 0.f, 0.f, 0.f, 0.f, 0.f, 0.f, 0.f};
      for (int kk = 0; kk < 8; ++kk) {
        const int kb = kk * 4 + koff;
        v2f a, b;
        a.x = As[(mt * 16 + l15) * LD + kb];
        a.y = As[(mt * 16 + l15) * LD + kb + 1];
        b.x = Bs[kb * ldb + nt * 16 + l15];
        b.y = Bs[(kb + 1) * ldb + nt * 16 + l15];
        acc = __builtin_amdgcn_wmma_f32_16x16x4_f32(false, a, false, b,
                                                    (short)0, acc, false, false);
      }
      for (int r = 0; r < 8; ++r) {
        const int orow = mt * 16 + r + half * 8;
        const int ocol = nt * 16 + l15;
        float v = acc[r] + addv[ocol];
        if (dotanh) v = tanhf(v);
        Ds[orow * LD + ocol] = v;
      }
    }
}

// vf(t_v, Y) -> K, factorized:  K = tanh(Y@W1 + b1) @ M_v + (b2 . dq_v)
// with M_v = W2 . dq_v precomputed for all 16 RK4 node times by one WMMA GEMM.
// Ysrc == Kdst is legal: gemm1 fully consumes Ysrc into Hs before gemm2
// overwrites Kdst.
__device__ __forceinline__
void vf_eval(int v, const float* Ysrc, float* Kdst,
             const float* B2s, const float* dqB, const float* W1s,
             const float* B1s, const float* Mall,
             float* Bt, float* Hs, int lane)
{
  {
    float acc = 0.f;
    for (int j = 0; j < DK; ++j) acc += B2s[lane * DK + j] * dqB[j * 16 + v];
    Bt[lane] = acc;                              // bt[i] = sum_j b2[i,j] dq[j]
  }
  wave_gemm_lds(Ysrc, W1s, LD, Hs, B1s, true, lane);   // hid = tanh(Y@W1 + b1)
  __syncthreads();
  wave_gemm_lds(Hs, Mall + v * (INTERN * DK), DK, Kdst, Bt, false, lane);
  __syncthreads();
}

// ---------------------------------------------------------------------------
// RK4 neural-CDE integration: one wave32 per (g, bh).
//  Phase 1: dq at all 16 RK4 node times (spline derivative).
//  Phase 2: M_all(1024x16) = W2(1024x32) @ dqB(32x16)  -- single-pass W2, WMMA.
//  Phase 3: 4 RK4 steps, each vf = two 40x32x32 WMMA GEMMs (in-place buffers).
//  Epilogue: seg_scores[g, bh, s] = (q_end . y_s) / sqrt(dk).
// ---------------------------------------------------------------------------
__global__ __launch_bounds__(32)
void NCDE_integrate(const float* __restrict__ kp,
                    const float* __restrict__ times,
                    const float* __restrict__ ca, const float* __restrict__ cb,
                    const float* __restrict__ cc, const float* __restrict__ cd,
                    const float* __restrict__ w1g, const float* __restrict__ b1g,
                    const float* __restrict__ w2g, const float* __restrict__ b2g,
                    float* __restrict__ segsc)
{
  __shared__ float Ys[48 * LD], K1[48 * LD], K2[48 * LD], K3[48 * LD], Hs[48 * LD];
  __shared__ float Mall[16 * INTERN * DK];   // 16 node times x M(32x32), stride 32
  __shared__ float dqB[DK * 16];             // dq[j, v]: B operand (32 x 16)
  __shared__ float W1s[32 * LD];
  __shared__ float B2s[DK * DK];
  __shared__ float Kn[SEQ], B1s[32], Bt[32], Qe[32];

  const int lane = threadIdx.x;
  const int g  = blockIdx.x >> 5;
  const int bh = blockIdx.x & 31;
  const int half = lane >> 4, l15 = lane & 15, koff = half * 2;

  __builtin_prefetch(w2g, 0, 3);   // -> global_prefetch_b8 (pull W2 toward cache)

  for (int i = lane; i < DK * DK; i += 32) {
    W1s[(i >> 5) * LD + (i & 31)] = w1g[i];
    B2s[i] = b2g[i];
  }
  for (int i = lane; i < SEQ; i += 32) Kn[i] = times[i];
  B1s[lane] = b1g[lane];
  for (int s = 0; s < SEQ; ++s) Ys[s * LD + lane] = kp[(bh * SEQ + s) * DK + lane];
  for (int s = SEQ; s < 48; ++s) Ys[s * LD + lane] = 0.f;
  __syncthreads();

  // segment times: chunk c, sub-segment jj
  const int c  = g >> 2;
  const int jj = g & 3;
  const float t_c0 = Kn[c * 5], t_c4 = Kn[c * 5 + 4];
  const float taustep = (t_c4 - t_c0) * 0.25f;   // /(NQ-1)
  const float t0seg = t_c0 + (float)jj * taustep;
  const float dt = taustep * 0.25f;              // /MSTEP

  // ---- Phase 1: dq at node time v = m*4+q  ->  t = t0seg + m*dt + q*dt/3
  for (int v = 0; v < 16; ++v) {
    const float tv = t0seg + (float)(v >> 2) * dt + (float)(v & 3) * (dt * (1.f / 3.f));
    int idx = 0;
    for (int i = 1; i < SEQ; ++i) if (tv >= Kn[i]) idx = i;
    if (idx > SEQ - 2) idx = SEQ - 2;
    const float s = tv - Kn[idx];
    const int co = idx * (BH * DK) + bh * DK + lane;     // lane == j
    dqB[lane * 16 + v] = cb[co] + s * (2.f * cc[co] + 3.f * s * cd[co]);
  }
  __syncthreads();

  // ---- Phase 2: M_all = W2(1024x32) @ dqB(32x16), single pass over W2
  for (int mt = 0; mt < 64; ++mt) {
    v8f acc = {0.f, 0.f, 0.f, 0.f, 0.f, 0.f, 0.f, 0.f};
    for (int kk = 0; kk < 8; ++kk) {
      const int kb = kk * 4 + koff;
      v2f a, b;
      a.x = w2g[(mt * 16 + l15) * DK + kb];
      a.y = w2g[(mt * 16 + l15) * DK + kb + 1];
      b.x = dqB[kb * 16 + l15];
      b.y = dqB[(kb + 1) * 16 + l15];
      acc = __builtin_amdgcn_wmma_f32_16x16x4_f32(false, a, false, b,
                                                  (short)0, acc, false, false);
    }
    for (int r = 0; r < 8; ++r) {
      const int row = mt * 16 + r + half * 8;            // row = h*32 + i
      Mall[l15 * (INTERN * DK) + row] = acc[r];          // col l15 == node v
    }
  }
  __syncthreads();

  // ---- Phase 3: RK4 (intermediate states built in place in K2/K3)
  for (int m = 0; m < 4; ++m) {
    const int v0 = m * 4;

    vf_eval(v0 + 0, Ys, K1, B2s, dqB, W1s, B1s, Mall, Bt, Hs, lane);   // k1
    for (int s = 0; s < SEQ; ++s) {
      const int o = s * LD + lane;
      K2[o] = Ys[o] + dt * K1[o] * (1.f / 3.f);
    }
    __syncthreads();

    vf_eval(v0 + 1, K2, K2, B2s, dqB, W1s, B1s, Mall, Bt, Hs, lane);   // k2
    for (int s = 0; s < SEQ; ++s) {
      const int o = s * LD + lane;
      K3[o] = Ys[o] + dt * (K2[o] - K1[o] * (1.f / 3.f));
    }
    __syncthreads();

    vf_eval(v0 + 2, K3, K3, B2s, dqB, W1s, B1s, Mall, Bt, Hs, lane);   // k3
    for (int s = 0; s < SEQ; ++s) {
      const int o = s * LD + lane;
      const float k1v = K1[o], k2v = K2[o], k3v = K3[o];
      K1[o] = k1v + 3.f * (k2v + k3v);             // k1 + 3(k2+k3)
      K3[o] = Ys[o] + dt * (k1v - k2v + k3v);      // y-input for k4
    }
    __syncthreads();

    vf_eval(v0 + 3, K3, K3, B2s, dqB, W1s, B1s, Mall, Bt, Hs, lane);   // k4
    for (int s = 0; s < SEQ; ++s) {
      const int o = s * LD + lane;
      Ys[o] += dt * (K1[o] + K3[o]) * 0.125f;
    }
    __syncthreads();
  }

  // ---- Epilogue: q_end = spline_eval(seg_t1); scores = (q_end . y_s)/sqrt(dk)
  const float t1 = t0seg + taustep;
  int idx = 0;
  for (int i = 1; i < SEQ; ++i) if (t1 >= Kn[i]) idx = i;
  if (idx > SEQ - 2) idx = SEQ - 2;
  const float sv = t1 - Kn[idx];
  const int co = idx * (BH * DK) + bh * DK + lane;
  Qe[lane] = ca[co] + sv * (cb[co] + sv * (cc[co] + sv * cd[co]));
  __syncthreads();

  const float rs = rsqrtf((float)DK);
  for (int s = lane; s < SEQ; s += 32) {
    float acc = 0.f;
    for (int c2 = 0; c2 < DK; ++c2) acc += Qe[c2] * Ys[s * LD + c2];
    segsc[(g * BH + bh) * SEQ + s] = acc * rs;
  }
}

// ---------------------------------------------------------------------------
// Chunk-average -> softmax -> attn @ V, one block per bh; 8 chunk-groups of
// 32 lanes. Rows within a chunk share the same attention row.
// ---------------------------------------------------------------------------
__global__ __launch_bounds__(256)
void NCDE_attn(const float* __restrict__ segsc, const float* __restrict__ vp,
               float* __restrict__ headout)
{
  __shared__ float cs[8][SEQ];
  __shared__ float wt[8][SEQ];
  __shared__ float mm[8], dd[8];
  const int bh = blockIdx.x;
  const int c = threadIdx.x >> 5, i = threadIdx.x & 31;
  const float rs = rsqrtf((float)DK);

  for (int s = i; s < SEQ; s += 32) {
    float a = 0.f;
    for (int j = 0; j < 4; ++j) a += segsc[((c * 4 + j) * BH + bh) * SEQ + s];
    cs[c][s] = a * 0.25f;
  }
  __syncthreads();
  if (i == 0) {
    float m = -3.4e38f;
    for (int s = 0; s < SEQ; ++s) { const float l = cs[c][s] * rs; if (l > m) m = l; }
    float d = 0.f;
    for (int s = 0; s < SEQ; ++s) d += expf(cs[c][s] * rs - m);
    mm[c] = m; dd[c] = d;
  }
  __syncthreads();
  for (int s = i; s < SEQ; s += 32) wt[c][s] = expf(cs[c][s] * rs - mm[c]) / dd[c];
  __syncthreads();

  float out = 0.f;
  for (int s = 0; s < SEQ; ++s) out += wt[c][s] * vp[(bh * SEQ + s) * DV + i];

  const int h = bh >> 2, b = bh & 3;   // bh = h*B + b, B = 4
  for (int r = 0; r < 5; ++r) {
    const int q = c * 5 + r;
    headout[(b * SEQ + q) * (NH * DV) + h * DV + i] = out;
  }
}

// ---------------------------------------------------------------------------
extern "C" void kernel_launch(void* const* d_in, const int* in_sizes, int n_in,
                              void* d_out, int out_size, void* d_ws, size_t ws_size,
                              hipStream_t stream)
{
  (void)in_sizes; (void)n_in; (void)out_size; (void)ws_size;
  const float* q     = (const float*)d_in[0];
  const float* k     = (const float*)d_in[1];
  const float* v     = (const float*)d_in[2];
  const float* times = (const float*)d_in[3];
  const float* w_q   = (const float*)d_in[4];
  const float* w_k   = (const float*)d_in[5];
  const float* w_v   = (const float*)d_in[6];
  const float* w_fc  = (const float*)d_in[7];
  const float* fk_w1 = (const float*)d_in[8];
  const float* fk_b1 = (const float*)d_in[9];
  const float* fk_w2 = (const float*)d_in[10];
  const float* fk_b2 = (const float*)d_in[11];

  float* ws      = (float*)d_ws;
  float* qp      = ws;                 // 40960  (BH, SEQ, DK)
  float* kp      = qp + 40960;         // 40960
  float* vp      = kp + 40960;         // 40960
  float* ca      = vp + 40960;         // 39936  (SEQ-1, BH*DK)
  float* cb      = ca + 39936;
  float* cc      = cb + 39936;
  float* cd      = cc + 39936;
  float* segsc   = cd + 39936;         // 40960  (G, BH, SEQ)
  float* headout = segsc + 40960;      // 40960  (B*SEQ, NH*DV)

  const dim3 tg(10, 16);               // 160/16 x 256/16 tiles
  NCDE_wmma_gemm<<<tg, 32, 0, stream>>>(q, w_q, qp, DMODEL, DMODEL, 1);
  NCDE_wmma_gemm<<<tg, 32, 0, stream>>>(k, w_k, kp, DMODEL, DMODEL, 1);
  NCDE_wmma_gemm<<<tg, 32, 0, stream>>>(v, w_v, vp, DMODEL, DMODEL, 1);
  NCDE_spline<<<4, 256, 0, stream>>>(qp, times, ca, cb, cc, cd);
  NCDE_integrate<<<GSEG * BH, 32, 0, stream>>>(kp, times, ca, cb, cc, cd,
                                               fk_w1, fk_b1, fk_w2, fk_b2, segsc);
  NCDE_attn<<<BH, 256, 0, stream>>>(segsc, vp, headout);
  NCDE_wmma_gemm<<<tg, 32, 0, stream>>>(headout, w_fc, (float*)d_out,
                                        DMODEL, DMODEL, 0);
}